// NormalizedGeometricAttention_34892314313426
// MI455X (gfx1250) — compile-verified
//
#include <hip/hip_runtime.h>
#include <math.h>

typedef __attribute__((ext_vector_type(2))) float v2f;
typedef __attribute__((ext_vector_type(4))) float v4f;
typedef __attribute__((ext_vector_type(8))) float v8f;

#define NB 8
#define NT 1024
#define NH 8
#define MVD 16
#define SCALE 0.25f
#define EPSV 1e-6f
#define LOG2E 1.44269504088896340736f
#define NQT 2   // query tiles (of 16 rows) per wave

// Cl(4,0) grade blocks: dims (1,4,6,4,1)
__device__ constexpr int GD[5]   = {1, 4, 6, 4, 1};
__device__ constexpr int GOFF[5] = {0, 1, 5, 11, 15};
__device__ constexpr int WOFF[5] = {0, 8, 136, 424, 552};  // cum H*d*d
__device__ constexpr int BOFF[5] = {0, 8, 40, 88, 120};    // cum H*d
#define WTOT 560
#define BTOT 128

struct ProjW { const float* p[6][5]; };  // rows: qw,qb,kw,kb,vw,vb ; cols: grade
struct OutW  { const float* ow[5]; const float* ob[5]; };

// ---------------------------------------------------------------------------
// Kernel 1: per-grade QKV projection; q/k normalized (cosine attention).
// V is written TRANSPOSED per head: [bh][d][T] so the attention PV A-operand
// loads are contiguous b64 in the hot loop.
// ---------------------------------------------------------------------------
__global__ __launch_bounds__(256) void k_proj(const float* __restrict__ x, ProjW W,
                                              float* __restrict__ qo,
                                              float* __restrict__ ko,
                                              float* __restrict__ vo) {
  __shared__ float sw[3 * WTOT + 3 * BTOT];
  float* swq = sw;            float* swk = sw + WTOT;  float* swv = sw + 2 * WTOT;
  float* sbq = sw + 3 * WTOT; float* sbk = sbq + BTOT; float* sbv = sbq + 2 * BTOT;
  const int tid = threadIdx.x;
  for (int g = 0; g < 5; ++g) {
    const int wn = NH * GD[g] * GD[g];
    for (int i = tid; i < wn; i += 256) {
      swq[WOFF[g] + i] = W.p[0][g][i];
      swk[WOFF[g] + i] = W.p[2][g][i];
      swv[WOFF[g] + i] = W.p[4][g][i];
    }
    const int bn = NH * GD[g];
    for (int i = tid; i < bn; i += 256) {
      sbq[BOFF[g] + i] = W.p[1][g][i];
      sbk[BOFF[g] + i] = W.p[3][g][i];
      sbv[BOFF[g] + i] = W.p[5][g][i];
    }
  }
  __syncthreads();

  const int gid = blockIdx.x * 256 + tid;
  const int h  = gid & (NH - 1);
  const int bt = gid >> 3;                 // b*NT + t
  const int b  = bt >> 10;
  const int t  = bt & (NT - 1);

  float xv[MVD];
  const float* xp = x + (size_t)bt * MVD;
#pragma unroll
  for (int i = 0; i < MVD; ++i) xv[i] = xp[i];

  float qv[MVD], kv[MVD], vv[MVD];
  float qs = 0.f, ks = 0.f;
#pragma unroll
  for (int g = 0; g < 5; ++g) {
    const int d = GD[g];
#pragma unroll
    for (int j = 0; j < d; ++j) {
      const int col = h * d + j;
      float aq = sbq[BOFF[g] + col];
      float ak = sbk[BOFF[g] + col];
      float av = sbv[BOFF[g] + col];
#pragma unroll
      for (int i = 0; i < d; ++i) {
        const float xi = xv[GOFF[g] + i];
        const int  wi = WOFF[g] + i * (NH * d) + col;
        aq = fmaf(xi, swq[wi], aq);
        ak = fmaf(xi, swk[wi], ak);
        av = fmaf(xi, swv[wi], av);
      }
      qv[GOFF[g] + j] = aq; kv[GOFF[g] + j] = ak; vv[GOFF[g] + j] = av;
      qs = fmaf(aq, aq, qs); ks = fmaf(ak, ak, ks);
    }
  }
  const float qi = 1.f / sqrtf(fmaxf(qs, EPSV));
  const float ki = 1.f / sqrtf(fmaxf(ks, EPSV));
  const int    bh    = b * NH + h;
  const size_t base  = ((size_t)bh * NT + t) * MVD;
  const size_t vbase = (size_t)bh * MVD * NT + t;        // V^T: [bh][d][T]
#pragma unroll
  for (int i = 0; i < MVD; ++i) {
    qo[base + i] = qv[i] * qi;
    ko[base + i] = kv[i] * ki;
    vo[vbase + (size_t)i * NT] = vv[i];
  }
}

// ---------------------------------------------------------------------------
// Kernel 2: single-pass attention, one wave32 per (b, h, 2x16 query rows).
// Scores are cosines * 0.25, i.e. bounded: softmax needs NO running max.
// S^T = K_tile x Q^T and O^T = V^T x P^T via chained v_wmma_f32_16x16x4_f32.
// SCALE*log2(e) folded into Q; p = exp2(score) directly (v_exp_f32).
// ---------------------------------------------------------------------------
__global__ __launch_bounds__(32) void k_attn(const float* __restrict__ Q,
                                             const float* __restrict__ K,
                                             const float* __restrict__ VT,
                                             float* __restrict__ O) {
  const int bh   = blockIdx.x >> 5;             // NT/(16*NQT) = 32 tile groups
  const int q0   = (blockIdx.x & 31) * (16 * NQT);
  const int lane = threadIdx.x;
  const int half = lane >> 4;
  const int r    = lane & 15;
  const bool h0  = (half == 0);

  const float* Qb = Q  + (size_t)bh * NT * MVD;
  const float* Kb = K  + (size_t)bh * NT * MVD;
  const float* Vb = VT + (size_t)bh * MVD * NT;

  // Q^T B-operand chunks: lane -> col = qrow = r, rows 4c+2*half .. +1
  v2f bq[NQT][4];
#pragma unroll
  for (int qq = 0; qq < NQT; ++qq)
#pragma unroll
    for (int c = 0; c < 4; ++c) {
      v2f tq = *(const v2f*)(Qb + (q0 + 16 * qq + r) * MVD + 4 * c + 2 * half);
      bq[qq][c].x = tq.x * (SCALE * LOG2E);
      bq[qq][c].y = tq.y * (SCALE * LOG2E);
    }

  v8f  acc[NQT];
  float Lrow[NQT];
#pragma unroll
  for (int qq = 0; qq < NQT; ++qq) {
    acc[qq] = (v8f){0.f, 0.f, 0.f, 0.f, 0.f, 0.f, 0.f, 0.f};
    Lrow[qq] = 0.f;
  }

  for (int k0 = 0; k0 < NT; k0 += 16) {
    if (k0 + 16 < NT) {  // uniform branch; warm next K/V tile
      __builtin_prefetch(Kb + (k0 + 16 + r) * MVD, 0, 1);
      __builtin_prefetch(Vb + (size_t)r * NT + k0 + 16, 0, 1);
    }
    // K tile A-operand (16 keys x 16 d): lane -> row=key=r, K elems 4c+2*half..+1
    v2f ak[4];
#pragma unroll
    for (int c = 0; c < 4; ++c)
      ak[c] = *(const v2f*)(Kb + (k0 + r) * MVD + 4 * c + 2 * half);

    // V^T A-operand (16 d x 16 keys): lane -> row=d=r, keys 4c+2*half..+1 (b64)
    v2f av[4];
#pragma unroll
    for (int c = 0; c < 4; ++c)
      av[c] = *(const v2f*)(Vb + (size_t)r * NT + k0 + 4 * c + 2 * half);

    v2f bp[NQT][4];
#pragma unroll
    for (int qq = 0; qq < NQT; ++qq) {
      v8f st = {0.f, 0.f, 0.f, 0.f, 0.f, 0.f, 0.f, 0.f};
#pragma unroll
      for (int c = 0; c < 4; ++c)
        st = __builtin_amdgcn_wmma_f32_16x16x4_f32(false, ak[c], false, bq[qq][c],
                                                   (short)0, st, false, false);
      // st[e] = log2-score of (key = e + 8*half, qrow = r); bounded, no max needed
      float p[8];
#pragma unroll
      for (int e = 0; e < 8; ++e) p[e] = __builtin_amdgcn_exp2f(st[e]);
      Lrow[qq] += ((p[0] + p[1]) + (p[2] + p[3])) + ((p[4] + p[5]) + (p[6] + p[7]));

      // P^T B-operand: lane needs P[qrow=r][keys 4c+2*half .. +1].
      // Select-before-shuffle: 4 half-wave exchanges cover the 4 remote pairs.
      const float r0 = __shfl_xor(h0 ? p[2] : p[0], 16, 32);
      const float r1 = __shfl_xor(h0 ? p[3] : p[1], 16, 32);
      const float r2 = __shfl_xor(h0 ? p[6] : p[4], 16, 32);
      const float r3 = __shfl_xor(h0 ? p[7] : p[5], 16, 32);
      bp[qq][0].x = h0 ? p[0] : r0;  bp[qq][0].y = h0 ? p[1] : r1;
      bp[qq][1].x = h0 ? p[4] : r2;  bp[qq][1].y = h0 ? p[5] : r3;
      bp[qq][2].x = h0 ? r0 : p[2];  bp[qq][2].y = h0 ? r1 : p[3];
      bp[qq][3].x = h0 ? r2 : p[6];  bp[qq][3].y = h0 ? r3 : p[7];
    }

#pragma unroll
    for (int qq = 0; qq < NQT; ++qq)
#pragma unroll
      for (int c = 0; c < 4; ++c)
        acc[qq] = __builtin_amdgcn_wmma_f32_16x16x4_f32(false, av[c], false, bp[qq][c],
                                                        (short)0, acc[qq], false, false);
  }

  // Normalize and store: O^T[d = e + 8*half][qrow = r] -> two float4 per lane
#pragma unroll
  for (int qq = 0; qq < NQT; ++qq) {
    const float Lfull = Lrow[qq] + __shfl_xor(Lrow[qq], 16, 32);
    const float inv = 1.f / Lfull;
    float* ob = O + ((size_t)bh * NT + q0 + 16 * qq + r) * MVD + half * 8;
    v4f o0, o1;
    o0.x = acc[qq][0] * inv; o0.y = acc[qq][1] * inv;
    o0.z = acc[qq][2] * inv; o0.w = acc[qq][3] * inv;
    o1.x = acc[qq][4] * inv; o1.y = acc[qq][5] * inv;
    o1.z = acc[qq][6] * inv; o1.w = acc[qq][7] * inv;
    *(v4f*)(ob)     = o0;
    *(v4f*)(ob + 4) = o1;
  }
}

// ---------------------------------------------------------------------------
// Kernel 3: per-grade output projection. One thread per (b,t).
// ---------------------------------------------------------------------------
__global__ __launch_bounds__(256) void k_out(const float* __restrict__ ho, OutW W,
                                             float* __restrict__ out) {
  __shared__ float sw[WTOT + MVD];
  const int tid = threadIdx.x;
  for (int g = 0; g < 5; ++g) {
    const int wn = NH * GD[g] * GD[g];
    for (int i = tid; i < wn; i += 256) sw[WOFF[g] + i] = W.ow[g][i];
    if (tid < GD[g]) sw[WTOT + GOFF[g] + tid] = W.ob[g][tid];
  }
  __syncthreads();

  const int bt = blockIdx.x * 256 + tid;   // b*NT + t
  const int b = bt >> 10, t = bt & (NT - 1);

  float o[MVD];
#pragma unroll
  for (int i = 0; i < MVD; ++i) o[i] = sw[WTOT + i];

  for (int h = 0; h < NH; ++h) {
    const float* hp = ho + ((size_t)(b * NH + h) * NT + t) * MVD;
    float hv[MVD];
#pragma unroll
    for (int i = 0; i < MVD; ++i) hv[i] = hp[i];
#pragma unroll
    for (int g = 0; g < 5; ++g) {
      const int d = GD[g];
#pragma unroll
      for (int dd = 0; dd < d; ++dd) {
        const float hx = hv[GOFF[g] + dd];
#pragma unroll
        for (int j = 0; j < d; ++j)
          o[GOFF[g] + j] = fmaf(hx, sw[WOFF[g] + (h * d + dd) * d + j], o[GOFF[g] + j]);
      }
    }
  }
  float* op = out + (size_t)bt * MVD;
#pragma unroll
  for (int i = 0; i < MVD; ++i) op[i] = o[i];
}

// ---------------------------------------------------------------------------
extern "C" void kernel_launch(void* const* d_in, const int* in_sizes, int n_in,
                              void* d_out, int out_size, void* d_ws, size_t ws_size,
                              hipStream_t stream) {
  const float* x = (const float*)d_in[0];
  ProjW pw; OutW ow;
  for (int g = 0; g < 5; ++g) {
    const int base = 1 + g * 8;   // qw,qb,kw,kb,vw,vb,ow,ob per grade
    pw.p[0][g] = (const float*)d_in[base + 0];
    pw.p[1][g] = (const float*)d_in[base + 1];
    pw.p[2][g] = (const float*)d_in[base + 2];
    pw.p[3][g] = (const float*)d_in[base + 3];
    pw.p[4][g] = (const float*)d_in[base + 4];
    pw.p[5][g] = (const float*)d_in[base + 5];
    ow.ow[g]   = (const float*)d_in[base + 6];
    ow.ob[g]   = (const float*)d_in[base + 7];
  }
  float* ws = (float*)d_ws;
  const size_t N = (size_t)NB * NH * NT * MVD;   // 1M floats per buffer
  float* qb = ws;
  float* kb = ws + N;
  float* vb = ws + 2 * N;     // stored transposed per head: [bh][d][T]
  float* hb = ws + 3 * N;

  hipLaunchKernelGGL(k_proj, dim3((NB * NT * NH) / 256), dim3(256), 0, stream,
                     x, pw, qb, kb, vb);
  hipLaunchKernelGGL(k_attn, dim3(NB * NH * (NT / (16 * NQT))), dim3(32), 0, stream,
                     qb, kb, vb, hb);
  hipLaunchKernelGGL(k_out, dim3((NB * NT) / 256), dim3(256), 0, stream,
                     hb, ow, (float*)d_out);
}